// MyNet_30176440221733
// MI455X (gfx1250) — compile-verified
//
#include <hip/hip_runtime.h>
#include <math.h>

#define N_NODES 100000
#define N_EDGES 1200000
#define RATIO_K 70000

typedef float v2f __attribute__((ext_vector_type(2)));
typedef float v8f __attribute__((ext_vector_type(8)));

struct SelState {
    unsigned hist[256];
    unsigned prefix;
    unsigned needed;
    unsigned tie;
    unsigned sel;
    float    sum;
};

// ---------------- generic fills ----------------
__global__ void fill_f32(float* p, float v, int n) {
    int i = blockIdx.x * blockDim.x + threadIdx.x;
    if (i < n) p[i] = v;
}
__global__ void fill_i32(int* p, int v, int n) {
    int i = blockIdx.x * blockDim.x + threadIdx.x;
    if (i < n) p[i] = v;
}

// ---------------- WMMA GEMM: C[M,NDIM] = A[M,KDIM] @ B[KDIM,NDIM] ----------------
// One wave per 16-row tile; M % 16 == 0, NDIM % 16 == 0.
// B is staged zero-padded into LDS once per block -> unconditional ds_load fragments.
// A fragments use clamped-address loads + value select (no EXEC-divergent loads).
// Uses V_WMMA_F32_16X16X4_F32 (full f32 precision, matches reference).
template <int KDIM, int NDIM>
__global__ void wmma_gemm(const float* __restrict__ A, const float* __restrict__ B,
                          float* __restrict__ C, int M) {
    constexpr int NT   = NDIM / 16;
    constexpr int KCH  = (KDIM + 3) / 4;
    constexpr int KPAD = KCH * 4;

    __shared__ float Bs[KPAD * NDIM];
    for (int idx = threadIdx.x; idx < KPAD * NDIM; idx += blockDim.x) {
        int kr = idx / NDIM, c = idx - kr * NDIM;
        Bs[idx] = (kr < KDIM) ? B[(size_t)kr * NDIM + c] : 0.f;
    }
    __syncthreads();

    int wave = (int)((blockIdx.x * blockDim.x + threadIdx.x) >> 5);
    int row0 = wave * 16;
    if (row0 >= M) return;              // wave-uniform; EXEC stays all-ones below
    int lane = threadIdx.x & 31;
    int m    = lane & 15;               // row within tile (A) / col within tile (B,C)
    int hi   = lane >> 4;               // half-wave selects K pair {0,1} vs {2,3}

    v8f zero = {0.f, 0.f, 0.f, 0.f, 0.f, 0.f, 0.f, 0.f};
    v8f acc[NT];
#pragma unroll
    for (int t = 0; t < NT; ++t) acc[t] = zero;

    const float* arow = A + (size_t)(row0 + m) * KDIM;
#pragma unroll
    for (int kk = 0; kk < KCH; ++kk) {
        int k0 = kk * 4 + hi * 2;
        // clamped-address loads (always in-bounds) + value select: no divergent branches
        float a0 = arow[(k0     < KDIM) ? k0       : 0];
        float a1 = arow[(k0 + 1 < KDIM) ? (k0 + 1) : 0];
        v2f a;
        a.x = (k0     < KDIM) ? a0 : 0.f;
        a.y = (k0 + 1 < KDIM) ? a1 : 0.f;
#pragma unroll
        for (int t = 0; t < NT; ++t) {
            int col = t * 16 + m;
            v2f b;
            b.x = Bs[k0 * NDIM + col];
            b.y = Bs[(k0 + 1) * NDIM + col];
            acc[t] = __builtin_amdgcn_wmma_f32_16x16x4_f32(
                false, a, false, b, (short)0, acc[t], false, false);
        }
    }
    // C/D layout: VGPR r -> lanes 0-15: (M=r, N=lane); lanes 16-31: (M=r+8, N=lane-16)
    int radd = hi * 8;
#pragma unroll
    for (int t = 0; t < NT; ++t) {
#pragma unroll
        for (int r = 0; r < 8; ++r) {
            C[(size_t)(row0 + r + radd) * NDIM + t * 16 + m] = acc[t][r];
        }
    }
}

// ---------------- GCN plumbing ----------------
__global__ void deg_edges(const int* __restrict__ src, const int* __restrict__ dst,
                          float* __restrict__ deg) {
    int e = blockIdx.x * blockDim.x + threadIdx.x;
    if (e < N_EDGES) atomicAdd(&deg[dst[e]], 1.0f);
}
__global__ void deg2_edges(const int* __restrict__ src, const int* __restrict__ dst,
                           const int* __restrict__ remap, float* __restrict__ deg) {
    int e = blockIdx.x * blockDim.x + threadIdx.x;
    if (e >= N_EDGES) return;
    int ns = remap[src[e]], nd = remap[dst[e]];
    if (ns >= 0 && nd >= 0) atomicAdd(&deg[nd], 1.0f);
}
__global__ void rsqrt_k(const float* __restrict__ deg, float* __restrict__ dinv, int n) {
    int i = blockIdx.x * blockDim.x + threadIdx.x;
    if (i < n) dinv[i] = rsqrtf(deg[i]);
}
__global__ void scatter_feat64(const int* __restrict__ src, const int* __restrict__ dst,
                               const float* __restrict__ h0, const float* __restrict__ dinv,
                               float* __restrict__ agg) {
    unsigned t = blockIdx.x * blockDim.x + threadIdx.x;
    if (t >= (unsigned)N_EDGES * 64u) return;
    int e = (int)(t >> 6), j = (int)(t & 63u);
    int s = src[e], d = dst[e];
    float w = dinv[s] * dinv[d];
    atomicAdd(&agg[(size_t)d * 64 + j], h0[(size_t)s * 64 + j] * w);
}
__global__ void conv1_final(const float* __restrict__ agg, const float* __restrict__ h0,
                            const float* __restrict__ dinv, const float* __restrict__ b1,
                            float* __restrict__ h) {
    int t = blockIdx.x * blockDim.x + threadIdx.x;
    if (t >= N_NODES * 64) return;
    int i = t >> 6, j = t & 63;
    float di = dinv[i];
    float v  = agg[t] + h0[t] * di * di + b1[j];
    h[t] = v > 0.f ? v : 0.f;
}
__global__ void dot64(const float* __restrict__ h, const float* __restrict__ Wp,
                      float* __restrict__ p0) {
    int i = blockIdx.x * blockDim.x + threadIdx.x;
    if (i >= N_NODES) return;
    const float* r = h + (size_t)i * 64;
    float s = 0.f;
#pragma unroll
    for (int j = 0; j < 64; ++j) s += r[j] * Wp[j];
    p0[i] = s;
}
__global__ void scatter_scalar(const int* __restrict__ src, const int* __restrict__ dst,
                               const float* __restrict__ p0, const float* __restrict__ dinv,
                               float* __restrict__ aggp) {
    int e = blockIdx.x * blockDim.x + threadIdx.x;
    if (e >= N_EDGES) return;
    int s = src[e], d = dst[e];
    atomicAdd(&aggp[d], p0[s] * dinv[s] * dinv[d]);
}
__global__ void score_kernel(const float* __restrict__ aggp, const float* __restrict__ p0,
                             const float* __restrict__ dinv, const float* __restrict__ bp,
                             float* __restrict__ score, unsigned* __restrict__ key) {
    int i = blockIdx.x * blockDim.x + threadIdx.x;
    if (i >= N_NODES) return;
    float di = dinv[i];
    float sc = tanhf(aggp[i] + p0[i] * di * di + bp[0]);
    score[i] = sc;
    unsigned ub = __float_as_uint(sc);
    key[i] = (ub & 0x80000000u) ? ~ub : (ub | 0x80000000u);  // monotone: larger key = larger f32
}

// ---------------- radix top-k select (k-th largest of 100000) ----------------
__global__ void select_init(SelState* st) {
    int t = threadIdx.x;
    if (t < 256) st->hist[t] = 0u;
    if (t == 0) {
        st->prefix = 0u;
        st->needed = (unsigned)RATIO_K;
        st->tie = 0u;
        st->sel = 0u;
        st->sum = 0.f;
    }
}
__global__ void hist_pass(const unsigned* __restrict__ key, SelState* st,
                          int shift, unsigned maskhi) {
    int i = blockIdx.x * blockDim.x + threadIdx.x;
    if (i >= N_NODES) return;
    unsigned k = key[i];
    if ((k & maskhi) == st->prefix) atomicAdd(&st->hist[(k >> shift) & 0xFFu], 1u);
}
__global__ void scan_pass(SelState* st, int shift) {
    unsigned needed = st->needed, c = 0u;
    int bsel = 0;
    for (int b = 255; b >= 0; --b) {
        unsigned hb = st->hist[b];
        if (c + hb >= needed) { bsel = b; break; }
        c += hb;
    }
    st->prefix |= ((unsigned)bsel) << shift;
    st->needed = needed - c;
    for (int b = 0; b < 256; ++b) st->hist[b] = 0u;
}
__global__ void select_nodes(const unsigned* __restrict__ key, const float* __restrict__ score,
                             SelState* st, int* __restrict__ remap, int* __restrict__ inv,
                             float* __restrict__ gate) {
    int i = blockIdx.x * blockDim.x + threadIdx.x;
    if (i >= N_NODES) return;
    unsigned k = key[i], T = st->prefix;
    bool take = false;
    if (k > T) take = true;
    else if (k == T) {
        unsigned t = atomicAdd(&st->tie, 1u);
        if (t < st->needed) take = true;
    }
    if (take) {
        int nw = (int)atomicAdd(&st->sel, 1u);
        remap[i] = nw;
        inv[nw]  = i;
        gate[nw] = score[i];
    }
}
__global__ void gather_pool(const float* __restrict__ h, const int* __restrict__ inv,
                            const float* __restrict__ gate, float* __restrict__ hp) {
    int t = blockIdx.x * blockDim.x + threadIdx.x;
    if (t >= RATIO_K * 64) return;
    int k = t >> 6, j = t & 63;
    hp[t] = h[(size_t)inv[k] * 64 + j] * gate[k];
}
__global__ void scatter_feat32(const int* __restrict__ src, const int* __restrict__ dst,
                               const int* __restrict__ remap, const float* __restrict__ h2p,
                               const float* __restrict__ dinv, float* __restrict__ agg) {
    unsigned t = blockIdx.x * blockDim.x + threadIdx.x;
    if (t >= (unsigned)N_EDGES * 32u) return;
    int e = (int)(t >> 5), j = (int)(t & 31u);
    int ns = remap[src[e]], nd = remap[dst[e]];
    if (ns < 0 || nd < 0) return;
    float w = dinv[ns] * dinv[nd];
    atomicAdd(&agg[(size_t)nd * 32 + j], h2p[(size_t)ns * 32 + j] * w);
}
__global__ void final_node(const float* __restrict__ agg, const float* __restrict__ h2p,
                           const float* __restrict__ dinv, const float* __restrict__ b2,
                           const float* __restrict__ Wfc1, const float* __restrict__ bfc1,
                           SelState* st) {
    __shared__ float red[256];
    int k = blockIdx.x * blockDim.x + threadIdx.x;
    float x1 = 0.f;
    if (k < RATIO_K) {
        float di  = dinv[k];
        float di2 = di * di;
        x1 = bfc1[0];
#pragma unroll
        for (int j = 0; j < 32; ++j) {
            float v = agg[(size_t)k * 32 + j] + h2p[(size_t)k * 32 + j] * di2 + b2[j];
            x1 += (v > 0.f ? v : 0.f) * Wfc1[j];
        }
    }
    red[threadIdx.x] = x1;
    __syncthreads();
    for (int s = 128; s > 0; s >>= 1) {
        if ((int)threadIdx.x < s) red[threadIdx.x] += red[threadIdx.x + s];
        __syncthreads();
    }
    if (threadIdx.x == 0) atomicAdd(&st->sum, red[0]);
}
__global__ void final_out(const float* Wfc2, const float* bfc2, const float* Wfc3,
                          const float* bfc3, const float* Wfc4, const float* bfc4,
                          const float* Wfc5, const float* bfc5, const float* ctv,
                          const float* csv, const float* alv, const float* cetv,
                          const SelState* st, float* out) {
    float ct = bfc2[0];
    for (int i = 0; i < 33; ++i) ct += ctv[i] * Wfc2[i];
    float cs = bfc3[0];
    for (int i = 0; i < 25; ++i) cs += csv[i] * Wfc3[i];
    float al = bfc4[0];
    for (int i = 0; i < 52; ++i) al += alv[i] * Wfc4[i];
    float ce = bfc5[0];
    for (int i = 0; i < 10; ++i) ce += cetv[i] * Wfc5[i];
    float catsum = ct + cs + al + ce;
    out[0] = st->sum * catsum / (4.0f * (float)RATIO_K);
}

// ---------------- launch ----------------
extern "C" void kernel_launch(void* const* d_in, const int* in_sizes, int n_in,
                              void* d_out, int out_size, void* d_ws, size_t ws_size,
                              hipStream_t stream) {
    const float* x    = (const float*)d_in[0];
    const int*   ei   = (const int*)d_in[1];
    const int*   src  = ei;
    const int*   dst  = ei + N_EDGES;
    const float* W1   = (const float*)d_in[2];
    const float* b1   = (const float*)d_in[3];
    const float* Wp   = (const float*)d_in[4];
    const float* bp   = (const float*)d_in[5];
    const float* W2   = (const float*)d_in[6];
    const float* b2   = (const float*)d_in[7];
    const float* Wfc1 = (const float*)d_in[8];
    const float* bfc1 = (const float*)d_in[9];
    const float* Wfc2 = (const float*)d_in[10];
    const float* bfc2 = (const float*)d_in[11];
    const float* Wfc3 = (const float*)d_in[12];
    const float* bfc3 = (const float*)d_in[13];
    const float* Wfc4 = (const float*)d_in[14];
    const float* bfc4 = (const float*)d_in[15];
    const float* Wfc5 = (const float*)d_in[16];
    const float* bfc5 = (const float*)d_in[17];
    const float* ctv  = (const float*)d_in[18];
    const float* csv  = (const float*)d_in[19];
    const float* alv  = (const float*)d_in[20];
    const float* cetv = (const float*)d_in[21];
    float* out = (float*)d_out;

    char* ws = (char*)d_ws;
    size_t off = 0;
    auto alloc = [&](size_t bytes) -> char* {
        char* p = ws + off;
        off += (bytes + 255) & ~(size_t)255;
        return p;
    };
    const size_t N64 = (size_t)N_NODES * 64 * sizeof(float);
    float*    h0    = (float*)alloc(N64);                       // later reused as hp [K,64]
    float*    agg1  = (float*)alloc(N64);                       // later: h2p [K,32] + agg2 [K,32]
    float*    h     = (float*)alloc(N64);
    float*    deg1  = (float*)alloc((size_t)N_NODES * 4);
    float*    dinv1 = (float*)alloc((size_t)N_NODES * 4);
    float*    p0    = (float*)alloc((size_t)N_NODES * 4);
    float*    aggp  = (float*)alloc((size_t)N_NODES * 4);
    float*    score = (float*)alloc((size_t)N_NODES * 4);
    unsigned* key   = (unsigned*)alloc((size_t)N_NODES * 4);
    int*      remap = (int*)alloc((size_t)N_NODES * 4);
    int*      inv   = (int*)alloc((size_t)RATIO_K * 4);
    float*    gate  = (float*)alloc((size_t)RATIO_K * 4);
    float*    deg2  = (float*)alloc((size_t)RATIO_K * 4);
    float*    dinv2 = (float*)alloc((size_t)RATIO_K * 4);
    SelState* st    = (SelState*)alloc(sizeof(SelState));
    // aliases (disjoint lifetimes)
    float* hp   = h0;                                           // [K,64]
    float* h2p  = agg1;                                         // [K,32]
    float* agg2 = agg1 + (size_t)RATIO_K * 32;                  // [K,32]

    const int B = 256;
    auto g = [&](long long n) { return (unsigned)((n + B - 1) / B); };
    auto gw = [&](int rows) {  // one wave per 16-row tile
        long long waves = rows / 16;
        return (unsigned)((waves * 32 + B - 1) / B);
    };

    // ---- conv1 ----
    fill_f32<<<g(N_NODES), B, 0, stream>>>(deg1, 1.0f, N_NODES);           // self loops
    deg_edges<<<g(N_EDGES), B, 0, stream>>>(src, dst, deg1);
    rsqrt_k<<<g(N_NODES), B, 0, stream>>>(deg1, dinv1, N_NODES);
    wmma_gemm<6, 64><<<gw(N_NODES), B, 0, stream>>>(x, W1, h0, N_NODES);
    fill_f32<<<g((long long)N_NODES * 64), B, 0, stream>>>(agg1, 0.f, N_NODES * 64);
    scatter_feat64<<<g((long long)N_EDGES * 64), B, 0, stream>>>(src, dst, h0, dinv1, agg1);
    conv1_final<<<g((long long)N_NODES * 64), B, 0, stream>>>(agg1, h0, dinv1, b1, h);

    // ---- pooling score ----
    dot64<<<g(N_NODES), B, 0, stream>>>(h, Wp, p0);
    fill_f32<<<g(N_NODES), B, 0, stream>>>(aggp, 0.f, N_NODES);
    scatter_scalar<<<g(N_EDGES), B, 0, stream>>>(src, dst, p0, dinv1, aggp);
    score_kernel<<<g(N_NODES), B, 0, stream>>>(aggp, p0, dinv1, bp, score, key);

    // ---- top-k radix select ----
    select_init<<<1, 256, 0, stream>>>(st);
    const int      shifts[4] = {24, 16, 8, 0};
    const unsigned masks[4]  = {0u, 0xFF000000u, 0xFFFF0000u, 0xFFFFFF00u};
    for (int p = 0; p < 4; ++p) {
        hist_pass<<<g(N_NODES), B, 0, stream>>>(key, st, shifts[p], masks[p]);
        scan_pass<<<1, 1, 0, stream>>>(st, shifts[p]);
    }
    fill_i32<<<g(N_NODES), B, 0, stream>>>(remap, -1, N_NODES);
    select_nodes<<<g(N_NODES), B, 0, stream>>>(key, score, st, remap, inv, gate);
    gather_pool<<<g((long long)RATIO_K * 64), B, 0, stream>>>(h, inv, gate, hp);

    // ---- conv2 on pooled graph ----
    wmma_gemm<64, 32><<<gw(RATIO_K), B, 0, stream>>>(hp, W2, h2p, RATIO_K);
    fill_f32<<<g(RATIO_K), B, 0, stream>>>(deg2, 1.0f, RATIO_K);
    deg2_edges<<<g(N_EDGES), B, 0, stream>>>(src, dst, remap, deg2);
    rsqrt_k<<<g(RATIO_K), B, 0, stream>>>(deg2, dinv2, RATIO_K);
    fill_f32<<<g((long long)RATIO_K * 32), B, 0, stream>>>(agg2, 0.f, RATIO_K * 32);
    scatter_feat32<<<g((long long)N_EDGES * 32), B, 0, stream>>>(src, dst, remap, h2p, dinv2, agg2);

    // ---- final reduction ----
    final_node<<<g(RATIO_K), B, 0, stream>>>(agg2, h2p, dinv2, b2, Wfc1, bfc1, st);
    final_out<<<1, 1, 0, stream>>>(Wfc2, bfc2, Wfc3, bfc3, Wfc4, bfc4, Wfc5, bfc5,
                                   ctv, csv, alv, cetv, st, out);
}